// DirSageConv_57432302682548
// MI455X (gfx1250) — compile-verified
//
#include <hip/hip_runtime.h>
#include <hip/hip_bf16.h>

typedef __attribute__((ext_vector_type(2))) float v2f;
typedef __attribute__((ext_vector_type(8))) float v8f;

#define D 128            // feature dim (D_IN == D_OUT == 128)
#define PAIR_STRIDE 144  // float2 row stride in LDS (bank-conflict-free: 2*144 % 64 == 32)

// ---------------------------------------------------------------------------
// Zero the workspace (agg_in, agg_out, cnt_in, cnt_out).
// ---------------------------------------------------------------------------
__global__ void zero_ws_kernel(float* __restrict__ ws, long total) {
    long i = (long)blockIdx.x * blockDim.x + threadIdx.x;
    long base = i * 4;
    if (base + 3 < total) {
        float4 z = make_float4(0.f, 0.f, 0.f, 0.f);
        *(float4*)(ws + base) = z;
    } else {
        for (long j = base; j < total; ++j) ws[j] = 0.f;
    }
}

// ---------------------------------------------------------------------------
// Edge scatter: one wave32 per edge. Each lane handles 4 contiguous floats
// (float4 gather, 512B coalesced per row). Hardware f32 atomics into the
// L2-resident accumulators; lane 0 bumps the degree counters.
// ---------------------------------------------------------------------------
__global__ void scatter_kernel(const float* __restrict__ x,
                               const int* __restrict__ ei,
                               float* __restrict__ aggin,
                               float* __restrict__ aggout,
                               float* __restrict__ cntin,
                               float* __restrict__ cntout,
                               int E) {
    int t = blockIdx.x * blockDim.x + threadIdx.x;
    int e = t >> 5;
    int lane = t & 31;
    if (e >= E) return;
    int src = ei[e];
    int dst = ei[E + e];
    const float4 xs = *(const float4*)(x + (size_t)src * D + lane * 4);
    const float4 xd = *(const float4*)(x + (size_t)dst * D + lane * 4);
    float* pi = aggin + (size_t)dst * D + lane * 4;
    float* po = aggout + (size_t)src * D + lane * 4;
    unsafeAtomicAdd(pi + 0, xs.x);
    unsafeAtomicAdd(pi + 1, xs.y);
    unsafeAtomicAdd(pi + 2, xs.z);
    unsafeAtomicAdd(pi + 3, xs.w);
    unsafeAtomicAdd(po + 0, xd.x);
    unsafeAtomicAdd(po + 1, xd.y);
    unsafeAtomicAdd(po + 2, xd.z);
    unsafeAtomicAdd(po + 3, xd.w);
    if (lane == 0) {
        unsafeAtomicAdd(cntin + dst, 1.0f);
        unsafeAtomicAdd(cntout + src, 1.0f);
    }
}

// ---------------------------------------------------------------------------
// Fused normalize + triple GEMM + bias using V_WMMA_F32_16X16X4_F32.
// Block = 256 threads = 8 waves; wave w owns rows [blk*128 + w*16, +16) and
// all 128 output columns (8 C tiles of 16x16 f32 = 64 VGPRs).
// Weights staged per-source into LDS as K-pair float2 rows:
//   lds[kk*PAIR_STRIDE + n] = { W[2kk][n], W[2kk+1][n] }
// so a B fragment is one ds_load_b64 and half-wave groups hit disjoint banks.
// ---------------------------------------------------------------------------
__global__ void __launch_bounds__(256)
gemm_kernel(const float* __restrict__ x,
            const float* __restrict__ Wself, const float* __restrict__ bself,
            const float* __restrict__ Ws2d,  const float* __restrict__ bs2d,
            const float* __restrict__ Wd2s,  const float* __restrict__ bd2s,
            const float* __restrict__ aggin, const float* __restrict__ aggout,
            const float* __restrict__ cntin, const float* __restrict__ cntout,
            float* __restrict__ out, int n) {
    extern __shared__ float2 lds[];  // 64 * PAIR_STRIDE float2 = 72 KB

    const int tid  = threadIdx.x;
    const int wave = tid >> 5;
    const int lane = tid & 31;
    const int m    = lane & 15;  // A row-in-tile / B+C column-in-tile
    const int g    = lane >> 4;  // half-wave K group
    const int rowbase = (blockIdx.x * 8 + wave) * 16;
    const int row  = rowbase + m;
    const bool rowok = row < n;
    const int rowc = rowok ? row : 0;

    v8f c[8];
#pragma unroll
    for (int nt = 0; nt < 8; ++nt) c[nt] = (v8f)0.0f;

    const float* srcs[3] = {x, aggin, aggout};
    const float* Ws[3]   = {Wself, Ws2d, Wd2s};
    float scl[3];
    scl[0] = 1.0f;
    scl[1] = 0.5f / fmaxf(cntin[rowc], 1.0f);   // (1-ALPHA) * mean
    scl[2] = 0.5f / fmaxf(cntout[rowc], 1.0f);  // ALPHA * mean

    for (int s = 0; s < 3; ++s) {
        __syncthreads();
        // Stage W_s into LDS as K-pair-interleaved float2 rows.
        const float* W = Ws[s];
        for (int i = tid; i < (D / 2) * D; i += 256) {
            int kk = i >> 7;   // 0..63
            int nn = i & 127;  // 0..127
            float2 p;
            p.x = W[(2 * kk) * D + nn];
            p.y = W[(2 * kk + 1) * D + nn];
            lds[kk * PAIR_STRIDE + nn] = p;
        }
        __syncthreads();

        const float* arow = srcs[s] + (size_t)rowc * D;
        const float sc = rowok ? scl[s] : 0.0f;
#pragma unroll 4
        for (int k = 0; k < D; k += 4) {
            float2 av = *(const float2*)(arow + k + 2 * g);
            v2f a;
            a.x = av.x * sc;
            a.y = av.y * sc;
            const int pr = (k >> 1) + g;  // pair row holding K = k+2g, k+2g+1
#pragma unroll
            for (int nt = 0; nt < 8; ++nt) {
                float2 bv = lds[pr * PAIR_STRIDE + nt * 16 + m];
                v2f b;
                b.x = bv.x;
                b.y = bv.y;
                c[nt] = __builtin_amdgcn_wmma_f32_16x16x4_f32(
                    false, a, false, b, (short)0, c[nt], false, false);
            }
        }
    }

    // bias + store: C tile VGPR r -> row rowbase + r + 8g, col nt*16 + m
#pragma unroll
    for (int nt = 0; nt < 8; ++nt) {
        const int col = nt * 16 + m;
        const float bias = bself[col] + 0.5f * bs2d[col] + 0.5f * bd2s[col];
#pragma unroll
        for (int r = 0; r < 8; ++r) {
            const int orow = rowbase + r + 8 * g;
            if (orow < n) out[(size_t)orow * D + col] = c[nt][r] + bias;
        }
    }
}

extern "C" void kernel_launch(void* const* d_in, const int* in_sizes, int n_in,
                              void* d_out, int out_size, void* d_ws, size_t ws_size,
                              hipStream_t stream) {
    const float* x     = (const float*)d_in[0];
    const float* Wself = (const float*)d_in[1];
    const float* bself = (const float*)d_in[2];
    const float* Ws2d  = (const float*)d_in[3];
    const float* bs2d  = (const float*)d_in[4];
    const float* Wd2s  = (const float*)d_in[5];
    const float* bd2s  = (const float*)d_in[6];
    const int*   ei    = (const int*)d_in[7];

    const int N = in_sizes[0] / D;
    const int E = in_sizes[7] / 2;

    float* ws     = (float*)d_ws;
    float* aggin  = ws;
    float* aggout = ws + (size_t)N * D;
    float* cntin  = ws + (size_t)2 * N * D;
    float* cntout = cntin + N;

    // 1) zero accumulators + counters
    const long total = (long)2 * N * D + 2L * N;
    const long c4 = (total + 3) / 4;
    zero_ws_kernel<<<(int)((c4 + 255) / 256), 256, 0, stream>>>(ws, total);

    // 2) edge scatter (one wave per edge)
    const long sthreads = (long)E * 32;
    scatter_kernel<<<(int)((sthreads + 255) / 256), 256, 0, stream>>>(
        x, ei, aggin, aggout, cntin, cntout, E);

    // 3) fused normalize + 3x GEMM + bias
    const int gblocks = (N + 127) / 128;
    const size_t shmem = (size_t)(D / 2) * PAIR_STRIDE * sizeof(float2);  // 72 KB
    gemm_kernel<<<gblocks, 256, shmem, stream>>>(
        x, Wself, bself, Ws2d, bs2d, Wd2s, bd2s,
        aggin, aggout, cntin, cntout, (float*)d_out, N);
}